// Idx2PixelLayer_31903017075374
// MI455X (gfx1250) — compile-verified
//
#include <hip/hip_runtime.h>

// Bilinear gather from a (2048, 2048, 8) f32 texture at 1M warped coordinates.
// Memory-bound gather: texture (128 MB) fits in MI455X's 192 MB L2, so we
// stream coords/output with non-temporal hints to preserve texture residency,
// and use 128-bit vector loads/stores everywhere (8 channels = 2 x b128).
// The Python floor-mod is computed with a branch-free floor/fma sequence
// (exact for the reference domain |x| < m) instead of libm fmodf, which
// otherwise expands to ~200 VALU/SALU instructions of serial latency.

typedef float f32x2 __attribute__((ext_vector_type(2)));
typedef float f32x4 __attribute__((ext_vector_type(4)));

#define TEX_H 2048
#define TEX_W 2048
#define TEX_C 8

__device__ __forceinline__ float floor_mod(float x, float m, float inv_m) {
    // Python/JAX semantics: result in [0, m), sign of divisor.
    // For |x| < m this is exactly: x < 0 ? x + m : x  (single rounding),
    // which matches jnp.mod's  lax.rem(x,m) (+ m on sign mismatch)  path.
    float k = floorf(x * inv_m);
    float r = fmaf(-k, m, x);
    // Fix-ups guard against off-by-one k from reciprocal rounding for
    // out-of-nominal |x| >= m inputs; dead code for the real data.
    if (r >= m)   r -= m;
    if (r < 0.0f) r += m;
    return r;
}

__global__ __launch_bounds__(256) void Idx2PixelLayer_kernel(
    const f32x2* __restrict__ coords,   // (N, 2)
    const f32x4* __restrict__ visible,  // (H, W, C) viewed as float4 pairs
    f32x4* __restrict__ out,            // (N, C) viewed as float4 pairs
    int n)
{
    int idx = blockIdx.x * blockDim.x + threadIdx.x;
    if (idx >= n) return;

    // Streaming read of this lane's coordinate pair (non-temporal: don't
    // pollute caches holding the texture).
    f32x2 xy = __builtin_nontemporal_load(coords + idx);

    const float m     = (float)(TEX_H - 4);          // 2044.0f (dims - 4)
    const float inv_m = 1.0f / (float)(TEX_H - 4);

    // c = (coord - 1.0) mod m + 1.0
    float c0 = floor_mod(xy.x - 1.0f, m, inv_m) + 1.0f;
    float c1 = floor_mod(xy.y - 1.0f, m, inv_m) + 1.0f;

    float f0 = floorf(c0);
    float f1 = floorf(c1);
    float d0 = c0 - f0;
    float d1 = c1 - f1;
    int   i0 = (int)f0;   // row,    in [1, 2044]
    int   i1 = (int)f1;   // column, in [1, 2044]

    // Pixel (r, cl) occupies float4 indices (r*W + cl)*2 and +1.
    int p_tl = (i0 * TEX_W + i1) * 2;   // g(0,0): visible[i0,   i1  ]
    int p_tr = p_tl + TEX_W * 2;        // g(1,0): visible[i0+1, i1  ]
    int p_bl = p_tl + 2;                // g(0,1): visible[i0,   i1+1]
    int p_br = p_tr + 2;                // g(1,1): visible[i0+1, i1+1]

    // 8 independent b128 gathers (L2-resident texture, default RT policy).
    f32x4 tl0 = visible[p_tl],     tl1 = visible[p_tl + 1];
    f32x4 tr0 = visible[p_tr],     tr1 = visible[p_tr + 1];
    f32x4 bl0 = visible[p_bl],     bl1 = visible[p_bl + 1];
    f32x4 br0 = visible[p_br],     br1 = visible[p_br + 1];

    // mid_bottom = br + d0*(bl - br); mid_top = tr + d0*(tl - tr);
    // out = mid_bottom + d1*(mid_top - mid_bottom)   (reference order)
    f32x4 mb0 = br0 + d0 * (bl0 - br0);
    f32x4 mb1 = br1 + d0 * (bl1 - br1);
    f32x4 mt0 = tr0 + d0 * (tl0 - tr0);
    f32x4 mt1 = tr1 + d0 * (tl1 - tr1);
    f32x4 r0  = mb0 + d1 * (mt0 - mb0);
    f32x4 r1  = mb1 + d1 * (mt1 - mb1);

    // Faithful reproduction of the (always-false) off-texture zeroing:
    // cond = (c0 > dims[0]) || (c0 > dims[0])
    if (c0 > (float)TEX_H) {
        r0 = (f32x4)0.0f;
        r1 = (f32x4)0.0f;
    }

    // Streaming output: non-temporal b128 stores keep the texture in L2.
    __builtin_nontemporal_store(r0, out + idx * 2);
    __builtin_nontemporal_store(r1, out + idx * 2 + 1);
}

extern "C" void kernel_launch(void* const* d_in, const int* in_sizes, int n_in,
                              void* d_out, int out_size, void* d_ws, size_t ws_size,
                              hipStream_t stream) {
    const f32x2* coords  = (const f32x2*)d_in[0];   // (N, 2) float32
    const f32x4* visible = (const f32x4*)d_in[1];   // (H, W, C) float32
    f32x4*       out     = (f32x4*)d_out;           // (N, C) float32

    int n = in_sizes[0] / 2;                        // number of points
    int threads = 256;
    int blocks  = (n + threads - 1) / threads;

    Idx2PixelLayer_kernel<<<blocks, threads, 0, stream>>>(coords, visible, out, n);
}